// RoIAlign_30940944400983
// MI455X (gfx1250) — compile-verified
//
#include <hip/hip_runtime.h>

// RoIAlign (PH=PW=7, SR=2, scale=1/16) for features [B=2,C=256,H=100,W=152] f32,
// rois [R,5] f32 -> out [R,C,7,7] f32.
//
// Design for MI455X (gfx1250, wave32):
//  - 1 workgroup per ROI, 256 threads = 8 waves; wave handles 2 channel-tiles of 16.
//  - Bilinear sampling in fp32 VALU (branchless, clamped, validity-masked).
//  - Pooling contraction out[c,bin] = sum_q 0.25*Sub[c,bin,q] done as
//    V_WMMA_F32_16X16X4_F32: M=16 channels, N=16 bins, K=4 sub-samples.
//    A layout (ISA 7.12.2): M=lane%16; lanes0-15 hold K=0,1; lanes16-31 hold K=2,3.
//    B rows are identical (0.25 in column t%16) -> robust to any B row striping.
//    D layout: lane L, vgpr r -> channel r+8*(L>=16), bin col L%16.

typedef float v2f __attribute__((ext_vector_type(2)));
typedef float v8f __attribute__((ext_vector_type(8)));

#define PHH 7
#define PWW 7
#define NBINS 49
#define RSCALE 0.0625f

__global__ __launch_bounds__(256) void roi_align_wmma_kernel(
    const float* __restrict__ features, const float* __restrict__ rois,
    float* __restrict__ out, int B, int C, int H, int W, int R)
{
    const int roi  = blockIdx.x;
    const int tid  = threadIdx.x;
    const int lane = tid & 31;
    const int wave = tid >> 5;
    if (roi >= R) return;

    // ---- ROI parameters (uniform across the block) ----
    const float* rp = rois + (size_t)roi * 5;
    const int   b      = (int)rp[0];
    const float roi_sw = rp[1] * RSCALE;
    const float roi_sh = rp[2] * RSCALE;
    const float roi_ew = rp[3] * RSCALE;
    const float roi_eh = rp[4] * RSCALE;
    const float roi_w  = fmaxf(roi_ew - roi_sw, 1.0f);
    const float roi_h  = fmaxf(roi_eh - roi_sh, 1.0f);
    const float bin_h  = roi_h * (1.0f / PHH);
    const float bin_w  = roi_w * (1.0f / PWW);

    const int m  = lane & 15;   // A-matrix M (channel in tile) == B/D column N
    const int iy = lane >> 4;   // sub-sample row: lanes<16 -> K{0,1}, lanes>=16 -> K{2,3}
    const int HW = H * W;
    const float fH = (float)H, fW = (float)W;

    for (int pass = 0; pass < 2; ++pass) {
        const int ctile = wave + pass * 8;          // 0..15
        const int c     = ctile * 16 + m;           // this lane's channel
        const float* f  = features + ((size_t)(b * C + c)) * (size_t)HW;

        v8f acc0 = {}; v8f acc1 = {}; v8f acc2 = {}; v8f acc3 = {};

        for (int t = 0; t < NBINS; ++t) {           // t = bin index (wave-uniform)
            const int ph = t / PWW;
            const int pw = t - ph * PWW;

            // y coordinate for this lane's sub-sample row (shared by its 2 samples)
            const float y  = roi_sh + ((float)ph + ((float)iy + 0.5f) * 0.5f) * bin_h;
            const bool  vy = (y >= -1.0f) && (y <= fH);
            const float yc = fminf(fmaxf(y, 0.0f), fH - 1.0f);
            const int   ylo = (int)floorf(yc);
            const int   yhi = (ylo + 1 < H) ? (ylo + 1) : (H - 1);
            const float ly  = yc - (float)ylo;
            const float hy  = 1.0f - ly;
            const float* frow0 = f + (size_t)ylo * W;
            const float* frow1 = f + (size_t)yhi * W;

            v2f a;
#pragma unroll
            for (int ix = 0; ix < 2; ++ix) {
                const float x  = roi_sw + ((float)pw + ((float)ix + 0.5f) * 0.5f) * bin_w;
                const bool  vx = (x >= -1.0f) && (x <= fW);
                const float xc = fminf(fmaxf(x, 0.0f), fW - 1.0f);
                const int   xlo = (int)floorf(xc);
                const int   xhi = (xlo + 1 < W) ? (xlo + 1) : (W - 1);
                const float lx  = xc - (float)xlo;
                const float hx  = 1.0f - lx;
                const float v = hy * (hx * frow0[xlo] + lx * frow0[xhi])
                              + ly * (hx * frow1[xlo] + lx * frow1[xhi]);
                const float s = (vy && vx) ? v : 0.0f;
                if (ix == 0) a.x = s; else a.y = s;
            }

            // B: 0.25 in column t%16 (all K rows identical -> layout-proof),
            // invalid samples already zeroed in A; mean divides by 4 regardless.
            const float bv = (m == (t & 15)) ? 0.25f : 0.0f;
            v2f bb; bb.x = bv; bb.y = bv;

            switch (t >> 4) {   // uniform branch: EXEC stays all-1s for WMMA
                case 0: acc0 = __builtin_amdgcn_wmma_f32_16x16x4_f32(
                            false, a, false, bb, (short)0, acc0, false, false); break;
                case 1: acc1 = __builtin_amdgcn_wmma_f32_16x16x4_f32(
                            false, a, false, bb, (short)0, acc1, false, false); break;
                case 2: acc2 = __builtin_amdgcn_wmma_f32_16x16x4_f32(
                            false, a, false, bb, (short)0, acc2, false, false); break;
                default: acc3 = __builtin_amdgcn_wmma_f32_16x16x4_f32(
                            false, a, false, bb, (short)0, acc3, false, false); break;
            }
        }

        // ---- Store: D layout -> lane L, vgpr r holds (channel r+8*(L>=16), bin L%16) ----
        const int chbase = ctile * 16 + 8 * iy;
#pragma unroll
        for (int nt = 0; nt < 4; ++nt) {
            const v8f acc = (nt == 0) ? acc0 : (nt == 1) ? acc1 : (nt == 2) ? acc2 : acc3;
            const int bin = nt * 16 + m;
            if (bin < NBINS) {
                float* op = out + ((size_t)roi * C + chbase) * (size_t)NBINS + bin;
#pragma unroll
                for (int r = 0; r < 8; ++r)
                    op[(size_t)r * NBINS] = acc[r];
            }
        }
    }
}

extern "C" void kernel_launch(void* const* d_in, const int* in_sizes, int n_in,
                              void* d_out, int out_size, void* d_ws, size_t ws_size,
                              hipStream_t stream) {
    const float* features = (const float*)d_in[0];
    const float* rois     = (const float*)d_in[1];
    float* out = (float*)d_out;

    // Shapes fixed by the reference setup: features [2,256,100,152], rois [R,5]
    const int B = 2, C = 256, H = 100, W = 152;
    const int R = in_sizes[1] / 5;

    roi_align_wmma_kernel<<<dim3(R), dim3(256), 0, stream>>>(
        features, rois, out, B, C, H, W, R);
}